// BaseProtonet_29222957482794
// MI455X (gfx1250) — compile-verified
//
#include <hip/hip_runtime.h>
#include <hip/hip_bf16.h>

typedef float v2f __attribute__((ext_vector_type(2)));
typedef float v8f __attribute__((ext_vector_type(8)));

#define NROWS   262144
#define DIMS    64
#define NPROTO  256
#define MSE_EPS 1e-5f

// One wave (32 lanes) computes a 16-row x 256-proto strip:
//   cross = F_tile(16x64) @ P^T(64x256) via 16 column-tiles of 16 protos,
//   each tile = 16 chained V_WMMA_F32_16X16X4_F32 over K=64.
// inv_d written directly; argmax fused via per-lane best + half-wave shfl tree.
__global__ __launch_bounds__(256) void protonet_wmma_kernel(
    const float* __restrict__ F,      // [NROWS, 64]
    const float* __restrict__ Pr,     // [NPROTO, 64]
    const int*   __restrict__ plab,   // [NPROTO]
    float*       __restrict__ inv_out,// [NROWS, NPROTO]
    int*         __restrict__ lab_out)// [NROWS]
{
    const int lane  = threadIdx.x & 31;
    const int wave  = threadIdx.x >> 5;
    const int lmod  = lane & 15;   // row-in-tile (A) / col-in-tile (B,C)
    const int lhalf = lane >> 4;   // 0: K pair {0,1}, 1: K pair {2,3}
    const long row0 = ((long)blockIdx.x * 8 + wave) * 16;

    // ---- Load A fragments (16x64 f32 tile) + row-norm partials ----
    // Lane layout per 16x16x4 f32 A frag: lane<16 -> M=lane, K={s4,s4+1};
    // lane>=16 -> M=lane-16, K={s4+2,s4+3}. Contiguous float2 per lane.
    const float* arow = F + (row0 + lmod) * DIMS + lhalf * 2;
    v2f a[16];
    float f2p = 0.0f;
#pragma unroll
    for (int s = 0; s < 16; ++s) {
        a[s] = *(const v2f*)(arow + s * 4);
        f2p += a[s].x * a[s].x + a[s].y * a[s].y;
    }
    // lane L and L^16 jointly hold all 64 K values of row (row0+lmod)
    float f2 = f2p + __shfl_xor(f2p, 16, 32); // ||f_{row0+lmod}||^2 in lanes L and L^16

    float best_val[8];
    int   best_idx[8];
#pragma unroll
    for (int i = 0; i < 8; ++i) { best_val[i] = -INFINITY; best_idx[i] = 0; }

#pragma unroll 1
    for (int j = 0; j < 16; ++j) {
        const int col = j * 16;
        // B[k,n] = Pr[n,k] -> identical float2 addressing pattern on proto rows.
        const float* brow = Pr + (long)(col + lmod) * DIMS + lhalf * 2;
        v8f acc = {0.f, 0.f, 0.f, 0.f, 0.f, 0.f, 0.f, 0.f};
        float p2p = 0.0f;
#pragma unroll
        for (int s = 0; s < 16; ++s) {
            v2f b = *(const v2f*)(brow + s * 4);
            p2p += b.x * b.x + b.y * b.y;
            // D = A(16x4) x B(4x16) + C, all f32
            acc = __builtin_amdgcn_wmma_f32_16x16x4_f32(
                /*neg_a=*/false, a[s], /*neg_b=*/false, b,
                /*c_mod=*/(short)0, acc, /*reuse_a=*/false, /*reuse_b=*/false);
        }
        float p2 = p2p + __shfl_xor(p2p, 16, 32); // ||p_{col+lmod}||^2

        // C/D layout: VGPR i -> lanes 0-15: (M=i, N=lane); lanes 16-31: (M=i+8, N=lane-16)
#pragma unroll
        for (int i = 0; i < 8; ++i) {
            const int m   = lhalf ? (i + 8) : i;
            const float f2m = __shfl(f2, m, 32);        // lane m (<16) holds f2 of row row0+m
            const float mse = (f2m - 2.0f * acc[i] + p2) * (1.0f / 64.0f);
            const float inv = 1.0f / (mse + MSE_EPS);
            inv_out[(row0 + m) * NPROTO + col + lmod] = inv;
            if (inv > best_val[i]) { best_val[i] = inv; best_idx[i] = col + lmod; }
        }
    }

    // ---- Fused argmax: reduce across the 16 lanes of each half-wave.
    // xor masks 1,2,4,8 never cross the half boundary (bit 4 untouched).
#pragma unroll
    for (int i = 0; i < 8; ++i) {
        float v  = best_val[i];
        int   idx = best_idx[i];
#pragma unroll
        for (int d = 8; d >= 1; d >>= 1) {
            float ov = __shfl_xor(v, d, 32);
            int   oi = __shfl_xor(idx, d, 32);
            // first-occurrence argmax: prefer strictly larger, or equal with lower index
            if (ov > v || (ov == v && oi < idx)) { v = ov; idx = oi; }
        }
        if (lmod == 0) {
            const int m = lhalf ? (i + 8) : i;
            lab_out[row0 + m] = plab[idx];
        }
    }
}

extern "C" void kernel_launch(void* const* d_in, const int* in_sizes, int n_in,
                              void* d_out, int out_size, void* d_ws, size_t ws_size,
                              hipStream_t stream) {
    (void)in_sizes; (void)n_in; (void)out_size; (void)d_ws; (void)ws_size;
    const float* F    = (const float*)d_in[0];   // features  [262144, 64] f32
    const float* Pr   = (const float*)d_in[1];   // prototypes [256, 64]   f32
    const int*   plab = (const int*)d_in[2];     // proto_labels [256]     i32

    float* inv_out = (float*)d_out;                                  // [N, 256] f32
    int*   lab_out = (int*)((float*)d_out + (size_t)NROWS * NPROTO); // [N] i32 (2nd tuple output)

    // 8 waves/block, each wave owns a 16-row tile -> 2048 blocks.
    const int blocks = NROWS / (8 * 16);
    protonet_wmma_kernel<<<blocks, 256, 0, stream>>>(F, Pr, plab, inv_out, lab_out);
}